// RoIAlign_4999341932727
// MI455X (gfx1250) — compile-verified
//
#include <hip/hip_runtime.h>
#include <hip/hip_bf16.h>

// RoIAlign forward, MI455X (gfx1250, wave32).
// Bandwidth-bound gather: ~176 MB compulsory HBM traffic (~7.5 us floor @23.3 TB/s).
// Feature map (124.5 MB) is L2-resident (192 MB global L2); gather re-reads hit L2.
// ROI record via scalar loads (blockIdx-derived uniform); corner pairs loaded as
// one 2xfp32 vector (8 gathers/thread instead of 16); output streamed with
// non-temporal stores so it does not evict the L2-resident feature map.

#define ROI_C      256
#define ROI_H      200
#define ROI_W      304
#define ROI_OUT    7
#define ROI_SCALE  0.25f

// 2xfp32 with true (4-byte) alignment: clang emits global_load_b64 when the
// target supports 4B-aligned b64 (ISA 9.5 DWORD auto-align), else two b32.
typedef float floatx2 __attribute__((ext_vector_type(2), aligned(4)));

__global__ __launch_bounds__(256)
void roi_align_fwd(const float* __restrict__ feat,   // (B, C, H, W)
                   const float* __restrict__ rois,   // (N, 5) [bidx, x1, y1, x2, y2]
                   float* __restrict__ out)          // (N, C, 7, 7)
{
    const int C = ROI_C, H = ROI_H, W = ROI_W;
    const int binsPerC   = ROI_OUT * ROI_OUT;       // 49
    const int blocksPerN = binsPerC;                // C*49/256 = 49 blocks per ROI

    // n derived from blockIdx.x only -> wave-uniform -> SGPR / s_load path.
    int n      = blockIdx.x / blocksPerN;
    int blkInN = blockIdx.x - n * blocksPerN;
    int rem    = blkInN * 256 + threadIdx.x;        // [0, 12544)
    int c      = rem / binsPerC;
    int bin    = rem - c * binsPerC;
    int ph     = bin / ROI_OUT;
    int pw     = bin - ph * ROI_OUT;

    // Uniform address: s_load_b128 + s_load_b32 (scalar cache, KMcnt).
    const float* r = rois + (size_t)n * 5;
    int   b  = (int)r[0];
    float sx = r[1] * ROI_SCALE;
    float sy = r[2] * ROI_SCALE;
    float ex = r[3] * ROI_SCALE;
    float ey = r[4] * ROI_SCALE;
    float rw = fmaxf(ex - sx, 1.0f);
    float rh = fmaxf(ey - sy, 1.0f);
    float bw = rw / (float)ROI_OUT;                 // true divide: match reference
    float bh = rh / (float)ROI_OUT;

    const float* plane = feat + ((size_t)(b * C + c)) * (size_t)(H * W);

    // ---- x-sample descriptors (depend on pw/ix only; hoisted out of iy) ----
    float lx[2], hx[2];
    int   xb[2];
    bool  xhi[2], vx[2];
    #pragma unroll
    for (int ix = 0; ix < 2; ++ix) {
        float xx = sx + (pw + 0.25f + 0.5f * (float)ix) * bw;
        vx[ix]  = (xx >= -1.0f) && (xx <= (float)W);
        float x = fminf(fmaxf(xx, 0.0f), (float)(W - 1));
        int  x0 = (int)floorf(x);
        lx[ix]  = x - (float)x0;
        hx[ix]  = 1.0f - lx[ix];
        xb[ix]  = min(x0, W - 2);                   // pair base: {row[xb], row[xb+1]}
        xhi[ix] = (x0 == W - 1);                    // border: both corners = pair.y
    }

    float acc = 0.0f;
    #pragma unroll
    for (int iy = 0; iy < 2; ++iy) {
        // sample rows: off = ph + 0.25, ph + 0.75  (matches (t%r + 0.5)/r)
        float yy = sy + (ph + 0.25f + 0.5f * (float)iy) * bh;
        bool  vy = (yy >= -1.0f) && (yy <= (float)H);
        float y  = fminf(fmaxf(yy, 0.0f), (float)(H - 1));
        int   y0 = (int)floorf(y);
        int   y1 = min(y0 + 1, H - 1);
        float ly = y - (float)y0;
        float hy = 1.0f - ly;
        const float* row0 = plane + (size_t)y0 * W;
        const float* row1 = plane + (size_t)y1 * W;
        #pragma unroll
        for (int ix = 0; ix < 2; ++ix) {
            floatx2 p0 = *(const floatx2*)(row0 + xb[ix]);
            floatx2 p1 = *(const floatx2*)(row1 + xb[ix]);
            float v00 = xhi[ix] ? p0.y : p0.x;      // border fixup (exact; lx==0 there)
            float v01 = p0.y;
            float v10 = xhi[ix] ? p1.y : p1.x;
            float v11 = p1.y;
            float v = hy * (hx[ix] * v00 + lx[ix] * v01)
                    + ly * (hx[ix] * v10 + lx[ix] * v11);
            if (vy && vx[ix]) acc += v;
        }
    }

    // Coalesced streamed store with non-temporal hint (preserve L2 for gathers).
    int outIdx = n * (C * binsPerC) + rem;          // ((n*C + c)*7 + ph)*7 + pw
    __builtin_nontemporal_store(acc * 0.25f, &out[outIdx]);
}

extern "C" void kernel_launch(void* const* d_in, const int* in_sizes, int n_in,
                              void* d_out, int out_size, void* d_ws, size_t ws_size,
                              hipStream_t stream)
{
    const float* feat = (const float*)d_in[0];   // (2, 256, 200, 304) fp32
    const float* rois = (const float*)d_in[1];   // (1024, 5) fp32
    float* out = (float*)d_out;                  // (1024, 256, 7, 7) fp32

    int N = in_sizes[1] / 5;                     // 1024
    int grid = N * ROI_OUT * ROI_OUT;            // 49 blocks per ROI = 50,176
    roi_align_fwd<<<grid, 256, 0, stream>>>(feat, rois, out);
}